// GAT_53489522704391
// MI455X (gfx1250) — compile-verified
//
#include <hip/hip_runtime.h>
#include <math.h>

#define N_NODES   10000
#define F_IN      128
#define HEADS     8
#define HIDD      256
#define HC        (HEADS * HIDD)     // 2048
#define NUM_CLASS 10
#define E_IN      50000
#define E_TOT     (E_IN + N_NODES)   // 60000 (self loops appended)
#define NEG_SLOPE 0.2f

typedef __attribute__((ext_vector_type(2))) float v2f;
typedef __attribute__((ext_vector_type(8))) float v8f;

// ---------------------------------------------------------------------------
// GEMM1: h1[N, 2048] = x[N,128] @ W1[128,2048], fp32 WMMA 16x16x4.
// Block = 256 threads (8 waves) -> 32x64 output tile, full K in LDS.
// ---------------------------------------------------------------------------
__launch_bounds__(256)
__global__ void gemm1_wmma(const float* __restrict__ X, const float* __restrict__ W,
                           float* __restrict__ Hout)
{
    __shared__ float As[32][F_IN + 4];   // +4 pad: kill stride-128 bank conflicts
    __shared__ float Bs[F_IN][64];

    const int tid = threadIdx.x;
    const int m0  = blockIdx.y * 32;
    const int n0  = blockIdx.x * 64;

    // Stage A (32x128) -- zero-fill rows past N_NODES
#pragma unroll
    for (int i = 0; i < 4; ++i) {
        int idx = (tid + i * 256) * 4;       // 4096 floats, float4 granules
        int r = idx >> 7;
        int c = idx & 127;
        float4 v = make_float4(0.f, 0.f, 0.f, 0.f);
        int gr = m0 + r;
        if (gr < N_NODES) v = *(const float4*)(X + (size_t)gr * F_IN + c);
        *(float4*)(&As[r][c]) = v;
    }
    // Stage B (128x64), coalesced float4
#pragma unroll
    for (int i = 0; i < 8; ++i) {
        int idx = (tid + i * 256) * 4;       // 8192 floats
        int r = idx >> 6;
        int c = idx & 63;
        *(float4*)(&Bs[r][c]) = *(const float4*)(W + (size_t)r * HC + n0 + c);
    }
    __syncthreads();

    const int wave = tid >> 5;
    const int lane = tid & 31;
    const int half = lane >> 4;      // K-half selector (ISA f32 A/B layout)
    const int ml   = lane & 15;
    const int ar   = (wave >> 2) * 16;   // wave's row offset in tile (0/16)
    const int bc   = (wave & 3) * 16;    // wave's col offset in tile (0..48)

    v8f acc = {0.f, 0.f, 0.f, 0.f, 0.f, 0.f, 0.f, 0.f};
#pragma unroll
    for (int kk = 0; kk < F_IN / 4; ++kk) {
        const int kb = kk * 4 + half * 2;
        v2f a, b;
        float2 av = *(const float2*)(&As[ar + ml][kb]);   // A[m][2*half+j]
        a.x = av.x; a.y = av.y;
        b.x = Bs[kb + 0][bc + ml];                        // B[2*half+j][n]
        b.y = Bs[kb + 1][bc + ml];
        acc = __builtin_amdgcn_wmma_f32_16x16x4_f32(false, a, false, b,
                                                    (short)0, acc, false, false);
    }

    const int col = n0 + bc + ml;
#pragma unroll
    for (int r = 0; r < 8; ++r) {
        const int row = m0 + ar + r + 8 * half;           // C/D layout
        if (row < N_NODES) Hout[(size_t)row * HC + col] = acc[r];
    }
}

// ---------------------------------------------------------------------------
// GEMM2: h2[N,10] = z[N,2048] @ W2[2048,10], fp32 WMMA, N-tile padded to 16.
// One wave per 16 rows; padded rows/cols clamped (independent per-column math).
// ---------------------------------------------------------------------------
__launch_bounds__(256)
__global__ void gemm2_wmma(const float* __restrict__ Z, const float* __restrict__ W2,
                           float* __restrict__ H2)
{
    const int tid  = threadIdx.x;
    const int wave = tid >> 5;
    const int lane = tid & 31;
    const int half = lane >> 4;
    const int ml   = lane & 15;
    const int m0   = (blockIdx.x * 8 + wave) * 16;
    const int ra   = min(m0 + ml, N_NODES - 1);      // clamp: never stored if OOB
    const int nb   = min(ml, NUM_CLASS - 1);         // clamp: cols 10..15 unused
    const float* zr = Z + (size_t)ra * HC;

    v8f acc = {0.f, 0.f, 0.f, 0.f, 0.f, 0.f, 0.f, 0.f};
#pragma unroll 4
    for (int k = 0; k < HC; k += 4) {
        const int kb = k + half * 2;
        v2f a, b;
        float2 av = *(const float2*)(zr + kb);
        a.x = av.x; a.y = av.y;
        b.x = W2[(size_t)(kb + 0) * NUM_CLASS + nb];
        b.y = W2[(size_t)(kb + 1) * NUM_CLASS + nb];
        acc = __builtin_amdgcn_wmma_f32_16x16x4_f32(false, a, false, b,
                                                    (short)0, acc, false, false);
    }
    if (ml < NUM_CLASS) {
#pragma unroll
        for (int r = 0; r < 8; ++r) {
            const int row = m0 + r + 8 * half;
            if (row < N_NODES) H2[(size_t)row * NUM_CLASS + ml] = acc[r];
        }
    }
}

// ---------------------------------------------------------------------------
// a_src/a_dst for layer 1: one wave per (node, head), 256-wide dot products.
// ---------------------------------------------------------------------------
__launch_bounds__(256)
__global__ void att1_kernel(const float* __restrict__ H1,
                            const float* __restrict__ att_src,
                            const float* __restrict__ att_dst,
                            float* __restrict__ aS, float* __restrict__ aD)
{
    const int wave = threadIdx.x >> 5;
    const int lane = threadIdx.x & 31;
    const int gw = blockIdx.x * 8 + wave;          // flat (n*HEADS + h)
    if (gw >= N_NODES * HEADS) return;
    const int n = gw / HEADS, hh = gw % HEADS;
    const float* hp = H1 + (size_t)n * HC + hh * HIDD;
    const float* as = att_src + hh * HIDD;
    const float* ad = att_dst + hh * HIDD;
    float ss = 0.f, sd = 0.f;
    for (int c = lane; c < HIDD; c += 32) {
        float v = hp[c];
        ss += v * as[c];
        sd += v * ad[c];
    }
    for (int off = 16; off > 0; off >>= 1) {
        ss += __shfl_down(ss, off, 32);
        sd += __shfl_down(sd, off, 32);
    }
    if (lane == 0) { aS[gw] = ss; aD[gw] = sd; }
}

// a_src/a_dst for layer 2: H=1, C=10 -> one thread per node.
__global__ void att2_kernel(const float* __restrict__ H2,
                            const float* __restrict__ as2,
                            const float* __restrict__ ad2,
                            float* __restrict__ aS, float* __restrict__ aD)
{
    const int n = blockIdx.x * blockDim.x + threadIdx.x;
    if (n >= N_NODES) return;
    float ss = 0.f, sd = 0.f;
#pragma unroll
    for (int c = 0; c < NUM_CLASS; ++c) {
        float v = H2[(size_t)n * NUM_CLASS + c];
        ss += v * as2[c];
        sd += v * ad2[c];
    }
    aS[n] = ss; aD[n] = sd;
}

// ---------------------------------------------------------------------------
// Edge machinery (shared between layers via runtime H).
// ---------------------------------------------------------------------------
__device__ __forceinline__ int2 edge_sd(const int* __restrict__ ei, int e)
{
    int s = (e < E_IN) ? ei[e]        : (e - E_IN);
    int d = (e < E_IN) ? ei[E_IN + e] : (e - E_IN);
    return make_int2(s, d);
}

// Float atomic max via sign-split integer atomics (monotone under IEEE order).
__device__ __forceinline__ void atomicMaxF(float* addr, float v)
{
    if (v >= 0.f) atomicMax((int*)addr, __float_as_int(v));
    else          atomicMin((unsigned int*)addr, __float_as_uint(v));
}

__global__ void edge_max(const int* __restrict__ ei,
                         const float* __restrict__ aS, const float* __restrict__ aD,
                         float* __restrict__ emax, int Hn)
{
    const int t = blockIdx.x * blockDim.x + threadIdx.x;
    if (t >= E_TOT * Hn) return;
    const int e = t / Hn, hh = t % Hn;
    const int2 sd = edge_sd(ei, e);
    float x = aS[sd.x * Hn + hh] + aD[sd.y * Hn + hh];
    x = (x >= 0.f) ? x : NEG_SLOPE * x;
    atomicMaxF(&emax[sd.y * Hn + hh], x);
}

__global__ void edge_expsum(const int* __restrict__ ei,
                            const float* __restrict__ aS, const float* __restrict__ aD,
                            const float* __restrict__ emax,
                            float* __restrict__ denom, float* __restrict__ exbuf, int Hn)
{
    const int t = blockIdx.x * blockDim.x + threadIdx.x;
    if (t >= E_TOT * Hn) return;
    const int e = t / Hn, hh = t % Hn;
    const int2 sd = edge_sd(ei, e);
    float x = aS[sd.x * Hn + hh] + aD[sd.y * Hn + hh];
    x = (x >= 0.f) ? x : NEG_SLOPE * x;
    float ex = expf(x - emax[sd.y * Hn + hh]);
    exbuf[t] = ex;
    atomicAdd(&denom[sd.y * Hn + hh], ex);
}

// Layer-1 aggregation: one 256-thread block per edge; 8 channels/thread.
__launch_bounds__(256)
__global__ void edge_agg1(const int* __restrict__ ei, const float* __restrict__ H1,
                          const float* __restrict__ exbuf, const float* __restrict__ denom,
                          float* __restrict__ agg)
{
    const int e  = blockIdx.x;
    const int2 sd = edge_sd(ei, e);
    const int cb = threadIdx.x * 8;        // channel base, all 8 in same head
    const int head = cb >> 8;              // / HIDD
    const float alpha = exbuf[(size_t)e * HEADS + head]
                      / (denom[sd.y * HEADS + head] + 1e-16f);
    const float4* hs = (const float4*)(H1 + (size_t)sd.x * HC + cb);
    float4 v0 = hs[0], v1 = hs[1];
    float* op = agg + (size_t)sd.y * HC + cb;
    atomicAdd(op + 0, alpha * v0.x); atomicAdd(op + 1, alpha * v0.y);
    atomicAdd(op + 2, alpha * v0.z); atomicAdd(op + 3, alpha * v0.w);
    atomicAdd(op + 4, alpha * v1.x); atomicAdd(op + 5, alpha * v1.y);
    atomicAdd(op + 6, alpha * v1.z); atomicAdd(op + 7, alpha * v1.w);
}

// Layer-2 aggregation: thread per (edge, class) (class padded to 16).
__global__ void edge_agg2(const int* __restrict__ ei, const float* __restrict__ H2,
                          const float* __restrict__ exbuf, const float* __restrict__ denom,
                          float* __restrict__ agg)
{
    const int t = blockIdx.x * blockDim.x + threadIdx.x;
    if (t >= E_TOT * 16) return;
    const int e = t >> 4, c = t & 15;
    if (c >= NUM_CLASS) return;
    const int2 sd = edge_sd(ei, e);
    const float alpha = exbuf[e] / (denom[sd.y] + 1e-16f);
    atomicAdd(&agg[(size_t)sd.y * NUM_CLASS + c],
              alpha * H2[(size_t)sd.x * NUM_CLASS + c]);
}

// ---------------------------------------------------------------------------
// Elementwise kernels
// ---------------------------------------------------------------------------
__global__ void fill_kernel(float* __restrict__ p, size_t n, float v)
{
    for (size_t i = blockIdx.x * (size_t)blockDim.x + threadIdx.x; i < n;
         i += (size_t)gridDim.x * blockDim.x)
        p[i] = v;
}

__global__ void bias_elu(const float* __restrict__ agg, const float* __restrict__ bias,
                         float* __restrict__ z)
{
    const size_t total = (size_t)N_NODES * HC;
    for (size_t i = blockIdx.x * (size_t)blockDim.x + threadIdx.x; i < total;
         i += (size_t)gridDim.x * blockDim.x) {
        float v = agg[i] + bias[i % HC];
        z[i] = (v > 0.f) ? v : expm1f(v);     // jax.nn.elu (alpha=1)
    }
}

__global__ void final_kernel(const float* __restrict__ agg2, const float* __restrict__ b2,
                             float* __restrict__ out)
{
    const int i = blockIdx.x * blockDim.x + threadIdx.x;
    if (i >= N_NODES * NUM_CLASS) return;
    float v = agg2[i] + b2[i % NUM_CLASS];
    out[i] = 1.f / (1.f + expf(-v));
}

// ---------------------------------------------------------------------------
extern "C" void kernel_launch(void* const* d_in, const int* in_sizes, int n_in,
                              void* d_out, int out_size, void* d_ws, size_t ws_size,
                              hipStream_t stream)
{
    const float* x   = (const float*)d_in[0];
    const int*   ei  = (const int*)d_in[1];      // [2,50000] int32, row0=src
    const float* W1  = (const float*)d_in[2];
    const float* as1 = (const float*)d_in[3];
    const float* ad1 = (const float*)d_in[4];
    const float* b1  = (const float*)d_in[5];
    const float* W2  = (const float*)d_in[6];
    const float* as2 = (const float*)d_in[7];
    const float* ad2 = (const float*)d_in[8];
    const float* b2  = (const float*)d_in[9];
    float* out = (float*)d_out;

    float* ws = (float*)d_ws;
    size_t off = 0;
    float* h1    = ws + off; off += (size_t)N_NODES * HC;      // also reused as z
    float* agg1  = ws + off; off += (size_t)N_NODES * HC;
    float* sA1   = ws + off; off += (size_t)N_NODES * HEADS;
    float* sD1   = ws + off; off += (size_t)N_NODES * HEADS;
    float* emax1 = ws + off; off += (size_t)N_NODES * HEADS;
    float* den1  = ws + off; off += (size_t)N_NODES * HEADS;
    float* ex1   = ws + off; off += (size_t)E_TOT * HEADS;
    float* h2    = ws + off; off += (size_t)N_NODES * NUM_CLASS;
    float* sA2   = ws + off; off += N_NODES;
    float* sD2   = ws + off; off += N_NODES;
    float* emax2 = ws + off; off += N_NODES;
    float* den2  = ws + off; off += N_NODES;
    float* ex2   = ws + off; off += E_TOT;
    float* agg2  = ws + off; off += (size_t)N_NODES * NUM_CLASS;
    (void)ws_size; (void)in_sizes; (void)n_in; (void)out_size;

    const dim3 b256(256);

    // ---- Layer 1 ----
    gemm1_wmma<<<dim3(HC / 64, (N_NODES + 31) / 32), b256, 0, stream>>>(x, W1, h1);
    att1_kernel<<<(N_NODES * HEADS) / 8, b256, 0, stream>>>(h1, as1, ad1, sA1, sD1);

    fill_kernel<<<4096, b256, 0, stream>>>(agg1, (size_t)N_NODES * HC, 0.f);
    fill_kernel<<<64,   b256, 0, stream>>>(den1, (size_t)N_NODES * HEADS, 0.f);
    fill_kernel<<<64,   b256, 0, stream>>>(emax1, (size_t)N_NODES * HEADS, -INFINITY);

    const int eg1 = (E_TOT * HEADS + 255) / 256;
    edge_max<<<eg1, b256, 0, stream>>>(ei, sA1, sD1, emax1, HEADS);
    edge_expsum<<<eg1, b256, 0, stream>>>(ei, sA1, sD1, emax1, den1, ex1, HEADS);
    edge_agg1<<<E_TOT, b256, 0, stream>>>(ei, h1, ex1, den1, agg1);

    bias_elu<<<4096, b256, 0, stream>>>(agg1, b1, h1);   // z overwrites h1

    // ---- Layer 2 ----
    gemm2_wmma<<<(N_NODES + 127) / 128, b256, 0, stream>>>(h1, W2, h2);
    att2_kernel<<<(N_NODES + 255) / 256, b256, 0, stream>>>(h2, as2, ad2, sA2, sD2);

    fill_kernel<<<8, b256, 0, stream>>>(den2, (size_t)N_NODES, 0.f);
    fill_kernel<<<8, b256, 0, stream>>>(emax2, (size_t)N_NODES, -INFINITY);
    fill_kernel<<<8, b256, 0, stream>>>(agg2, (size_t)N_NODES * NUM_CLASS, 0.f);

    const int eg2 = (E_TOT + 255) / 256;
    edge_max<<<eg2, b256, 0, stream>>>(ei, sA2, sD2, emax2, 1);
    edge_expsum<<<eg2, b256, 0, stream>>>(ei, sA2, sD2, emax2, den2, ex2, 1);
    edge_agg2<<<(E_TOT * 16 + 255) / 256, b256, 0, stream>>>(ei, h2, ex2, den2, agg2);

    final_kernel<<<(N_NODES * NUM_CLASS + 255) / 256, b256, 0, stream>>>(agg2, b2, out);
}